// retina_loss_37314675867688
// MI455X (gfx1250) — compile-verified
//
#include <hip/hip_runtime.h>
#include <hip/hip_bf16.h>
#include <math.h>

#define B_ 8
#define A_ 49104
#define M_ 50
#define K_ 80
#define BETA_ (1.0f / 9.0f)

typedef float v2f __attribute__((ext_vector_type(2)));
typedef float v8f __attribute__((ext_vector_type(8)));

// ---------------------------------------------------------------------------
// Block reduction (blockDim.x == 256) whose final 64->1 step is done with
// V_WMMA_F32_16X16X4_F32 against an all-ones B matrix: D = A * 1 yields row
// sums; summing a lane's 8 accumulator VGPRs gives half a column, and a
// shfl_xor(16) pairs lanes L / L+16 to produce the full 16-row total.
// Result valid on thread 0.
// ---------------------------------------------------------------------------
__device__ __forceinline__ float block_reduce_wmma(float v, float* red) {
    const int tid = threadIdx.x;
    __syncthreads();            // protect shared buffer reuse
    red[tid] = v;
    __syncthreads();
    if (tid < 64) {
        red[tid] = red[tid] + red[tid + 64] + red[tid + 128] + red[tid + 192];
    }
    __syncthreads();
    float total = 0.0f;
    if (tid < 32) {             // exactly wave 0 on wave32: EXEC all ones
#if defined(__gfx1250__)
        v2f a; a.x = red[tid]; a.y = red[tid + 32];   // 64 partials -> A(16x4)
        v2f b; b.x = 1.0f; b.y = 1.0f;                // B(4x16) = ones
        v8f c = {};
        c = __builtin_amdgcn_wmma_f32_16x16x4_f32(
                /*neg_a=*/false, a, /*neg_b=*/false, b,
                /*c_mod=*/(short)0, c, /*reuse_a=*/false, /*reuse_b=*/false);
        float s = c[0] + c[1] + c[2] + c[3] + c[4] + c[5] + c[6] + c[7];
        s += __shfl_xor(s, 16, 32);                   // rows 0-7 + rows 8-15
        total = s;
#else
        float s = red[tid] + red[tid + 32];
        for (int off = 16; off > 0; off >>= 1) s += __shfl_down(s, off, 32);
        total = s;
#endif
    }
    return total;
}

__device__ __forceinline__ float smooth_l1(float d) {
    d = fabsf(d);
    return (d <= BETA_) ? (0.5f * d * d / BETA_) : (d - 0.5f * BETA_);
}

// ---------------------------------------------------------------------------
// Kernel 0: zero the 24 accumulator words (posSum[8], regSum[8], clsSum[8]).
// ---------------------------------------------------------------------------
__global__ void rl_init_kernel(float* acc) {
    if (threadIdx.x < 24) acc[threadIdx.x] = 0.0f;
}

// ---------------------------------------------------------------------------
// Kernel 1: anchor assignment + regression loss.  grid = (ceil(A/256), B)
// ---------------------------------------------------------------------------
__global__ void rl_assign_kernel(const float* __restrict__ reg_preds,
                                 const float* __restrict__ anchors,
                                 const float* __restrict__ boxes,
                                 const int*   __restrict__ classes,
                                 int*   __restrict__ meta,
                                 float* __restrict__ posSum,
                                 float* __restrict__ regSum) {
    const int b   = blockIdx.y;
    const int a   = blockIdx.x * blockDim.x + threadIdx.x;
    const int tid = threadIdx.x;

    __shared__ float sb[M_ * 4];
    __shared__ int   scls[M_];
    __shared__ float red[256];

    if (tid < M_ * 4) sb[tid]  = boxes[b * M_ * 4 + tid];
    if (tid < M_)     scls[tid] = classes[b * M_ + tid];
    __syncthreads();

    float posf = 0.0f, rsum = 0.0f;
    if (a < A_) {
        const float4 an = ((const float4*)anchors)[a];
        const float aw = an.z - an.x, ah = an.w - an.y;
        const float areaA = aw * ah;

        float best = -2.0f; int bestIdx = 0;
        #pragma unroll 5
        for (int m = 0; m < M_; ++m) {
            const float bx0 = sb[m * 4 + 0], by0 = sb[m * 4 + 1];
            const float bx1 = sb[m * 4 + 2], by1 = sb[m * 4 + 3];
            float iw = fmaxf(fminf(an.z, bx1) - fmaxf(an.x, bx0), 0.0f);
            float ih = fmaxf(fminf(an.w, by1) - fmaxf(an.y, by0), 0.0f);
            const float inter = iw * ih;
            const float areaB = (bx1 - bx0) * (by1 - by0);
            const float ua    = fmaxf(areaA + areaB - inter, 1e-8f);
            float iou = inter / ua;
            if (bx0 == -1.0f) iou = -1.0f;          // padded (invalid) box
            if (iou > best) { best = iou; bestIdx = m; }  // argmax-first
        }
        const bool pos = best >= 0.5f;
        const bool neg = best <  0.4f;
        const int  c   = scls[bestIdx] - 1;          // 0..K-1
        const int  flag = pos ? 2 : (neg ? 0 : 1);
        meta[b * A_ + a] = (flag << 8) | (c & 255);

        if (pos) {
            posf = 1.0f;
            const float g0 = sb[bestIdx * 4 + 0], g1 = sb[bestIdx * 4 + 1];
            const float g2 = sb[bestIdx * 4 + 2], g3 = sb[bestIdx * 4 + 3];
            const float gw0 = g2 - g0, gh0 = g3 - g1;
            const float gcx = g0 + 0.5f * gw0, gcy = g1 + 0.5f * gh0;
            const float gw = fmaxf(gw0, 1.0f), gh = fmaxf(gh0, 1.0f);
            const float acx = an.x + 0.5f * aw, acy = an.y + 0.5f * ah;
            const float t0 = ((gcx - acx) / aw) / 0.1f;
            const float t1 = ((gcy - acy) / ah) / 0.1f;
            const float t2 = logf(gw / aw) / 0.2f;
            const float t3 = logf(gh / ah) / 0.2f;
            const float4 r = ((const float4*)reg_preds)[b * A_ + a];
            rsum = smooth_l1(t0 - r.x) + smooth_l1(t1 - r.y) +
                   smooth_l1(t2 - r.z) + smooth_l1(t3 - r.w);
        }
    }

    const float pTot = block_reduce_wmma(posf, red);
    const float rTot = block_reduce_wmma(rsum, red);
    if (tid == 0) {
        atomicAdd(&posSum[b], pTot);
        atomicAdd(&regSum[b], rTot);
    }
}

// ---------------------------------------------------------------------------
// Kernel 2: focal loss, single streaming pass over cls_logits (126 MB).
// grid = (1024, B).  Each float4 lies inside one anchor row (80 % 4 == 0).
// ---------------------------------------------------------------------------
__global__ void rl_focal_kernel(const float* __restrict__ logits,
                                const int*   __restrict__ meta,
                                float* __restrict__ clsSum) {
    const int b = blockIdx.y;
    __shared__ float red[256];

    const int N4 = A_ * K_ / 4;
    const float4* __restrict__ lg4 = (const float4*)(logits + (size_t)b * A_ * K_);
    const int*    __restrict__ mb  = meta + b * A_;

    float sum = 0.0f;
    const int stride = gridDim.x * blockDim.x;
    for (int i = blockIdx.x * blockDim.x + threadIdx.x; i < N4; i += stride) {
        const float4 p4 = lg4[i];
        if (i + stride < N4) __builtin_prefetch(lg4 + i + stride, 0, 1);

        const int e   = i * 4;
        const int anc = e / K_;
        const int k0  = e - anc * K_;
        const int m    = mb[anc];
        const int flag = m >> 8;
        if (flag != 1) {                              // ignore anchors: 0
            const int  c     = m & 255;
            const bool isPos = (flag == 2);
            const float pp[4] = {p4.x, p4.y, p4.z, p4.w};
            #pragma unroll
            for (int j = 0; j < 4; ++j) {
                const float p = fminf(fmaxf(pp[j], 1e-4f), 1.0f - 1e-4f);
                if (isPos && (k0 + j) == c)
                    sum += 0.25f * (1.0f - p) * (1.0f - p) * (-logf(p));
                else
                    sum += 0.75f * p * p * (-log1pf(-p));
            }
        }
    }

    const float tot = block_reduce_wmma(sum, red);
    if (threadIdx.x == 0) atomicAdd(&clsSum[b], tot);
}

// ---------------------------------------------------------------------------
// Kernel 3: per-image normalization + batch mean -> 3 scalars.
// ---------------------------------------------------------------------------
__global__ void rl_finalize_kernel(const float* __restrict__ posSum,
                                   const float* __restrict__ regSum,
                                   const float* __restrict__ clsSum,
                                   float* __restrict__ out) {
    if (threadIdx.x == 0 && blockIdx.x == 0) {
        float cl = 0.0f, rl = 0.0f;
        for (int b = 0; b < B_; ++b) {
            const float np = fmaxf(posSum[b], 1.0f);
            cl += clsSum[b] / np;
            rl += regSum[b] / (4.0f * np);
        }
        cl *= (1.0f / B_);
        rl *= (1.0f / B_);
        out[0] = cl;
        out[1] = rl;
        out[2] = cl + rl;
    }
}

// ---------------------------------------------------------------------------
extern "C" void kernel_launch(void* const* d_in, const int* in_sizes, int n_in,
                              void* d_out, int out_size, void* d_ws, size_t ws_size,
                              hipStream_t stream) {
    const float* cls_logits = (const float*)d_in[0];   // (B, A, K) f32
    const float* reg_preds  = (const float*)d_in[1];   // (B, A, 4) f32
    const float* anchors    = (const float*)d_in[2];   // (A, 4)    f32
    const float* boxes      = (const float*)d_in[3];   // (B, M, 4) f32
    const int*   classes    = (const int*)  d_in[4];   // (B, M)    i32
    float* out = (float*)d_out;

    // Workspace layout: [posSum f32 x8][regSum f32 x8][clsSum f32 x8] @0,
    // meta i32 x (B*A) @128 bytes.
    float* acc    = (float*)d_ws;
    float* posSum = acc;
    float* regSum = acc + 8;
    float* clsSum = acc + 16;
    int*   meta   = (int*)((char*)d_ws + 128);

    rl_init_kernel<<<1, 32, 0, stream>>>(acc);

    dim3 gridA((A_ + 255) / 256, B_);
    rl_assign_kernel<<<gridA, 256, 0, stream>>>(reg_preds, anchors, boxes,
                                                classes, meta, posSum, regSum);

    dim3 gridF(1024, B_);
    rl_focal_kernel<<<gridF, 256, 0, stream>>>(cls_logits, meta, clsSum);

    rl_finalize_kernel<<<1, 32, 0, stream>>>(posSum, regSum, clsSum, out);
}